// LogicLayer_16776142258879
// MI455X (gfx1250) — compile-verified
//
#include <hip/hip_runtime.h>

// out(4096x8192) = c0 + cA*A + cB*B + cAB*A*B  (per-row coeffs from table softmax)
// A = softmax_rows(Wa) @ prev, B = softmax_rows(Wb) @ prev, prev (1024x8192).
// Dual f32 GEMM fused on shared B-operand, V_WMMA_F32_16X16X4_F32.
// Block tile: M=128 (8 waves x 16 rows), N=128. prev staged transposed in LDS,
// double-buffered and software-pipelined (prefetch chunk c+1 during compute c).

typedef __attribute__((ext_vector_type(2))) float v2f;
typedef __attribute__((ext_vector_type(8))) float v8f;

#if defined(__HIP_DEVICE_COMPILE__) && defined(__gfx1250__)
#if !__has_builtin(__builtin_amdgcn_wmma_f32_16x16x4_f32)
#error "missing __builtin_amdgcn_wmma_f32_16x16x4_f32 on this toolchain (device)"
#endif
#endif

#define K_DIM 1024
#define N_DIM 8192
#define M_DIM 4096

#define BM 128       // block M tile (8 waves x 16)
#define BN 128       // block N tile
#define KC 16        // K chunk staged in LDS
#define LSTR 20      // LDS row stride in floats (16 K + 4 pad: alignment + banks)
#define NCHUNK (K_DIM / KC)

// ---------------------------------------------------------------------------
// Kernel 1: row softmax over 1024 elements. One block (256 threads) per row.
// ---------------------------------------------------------------------------
__global__ __launch_bounds__(256) void row_softmax_kernel(
    const float* __restrict__ w, float* __restrict__ p) {
  __shared__ float red[256];
  const int row = blockIdx.x;
  const int t = threadIdx.x;
  const float4* src = (const float4*)(w + (size_t)row * K_DIM);
  float4* dst = (float4*)(p + (size_t)row * K_DIM);

  float4 v = src[t];  // 256 threads * 4 floats = 1024
  float m = fmaxf(fmaxf(v.x, v.y), fmaxf(v.z, v.w));
  red[t] = m;
  __syncthreads();
#pragma unroll
  for (int s = 128; s > 0; s >>= 1) {
    if (t < s) red[t] = fmaxf(red[t], red[t + s]);
    __syncthreads();
  }
  m = red[0];
  __syncthreads();

  float4 e;
  e.x = __expf(v.x - m);
  e.y = __expf(v.y - m);
  e.z = __expf(v.z - m);
  e.w = __expf(v.w - m);
  red[t] = e.x + e.y + e.z + e.w;
  __syncthreads();
#pragma unroll
  for (int s = 128; s > 0; s >>= 1) {
    if (t < s) red[t] += red[t + s];
    __syncthreads();
  }
  const float inv = 1.0f / red[0];
  e.x *= inv; e.y *= inv; e.z *= inv; e.w *= inv;
  dst[t] = e;
}

// ---------------------------------------------------------------------------
// Kernel 2: 16-way softmax over table columns, folded into 4 coefficients.
// ---------------------------------------------------------------------------
__global__ __launch_bounds__(256) void table_coeff_kernel(
    const float* __restrict__ tw, float4* __restrict__ coef) {
  const int n = blockIdx.x * 256 + threadIdx.x;
  if (n >= M_DIM) return;
  float x[16];
  float m = -3.4e38f;
#pragma unroll
  for (int k = 0; k < 16; ++k) {
    x[k] = tw[(size_t)k * M_DIM + n];
    m = fmaxf(m, x[k]);
  }
  float s = 0.0f;
#pragma unroll
  for (int k = 0; k < 16; ++k) {
    x[k] = __expf(x[k] - m);
    s += x[k];
  }
  const float inv = 1.0f / s;
#pragma unroll
  for (int k = 0; k < 16; ++k) x[k] *= inv;

  float4 c;
  c.y = x[2] + x[3] + x[6] + x[7] - x[8] - x[9] - x[12] - x[13];          // cA
  c.z = x[4] + x[5] + x[6] + x[7] - x[8] - x[9] - x[10] - x[11];          // cB
  c.w = x[1] - x[2] - x[4] - 2.0f * x[6] - x[7] + x[8] + 2.0f * x[9] +
        x[11] + x[13] - x[14];                                            // cAB
  c.x = x[8] + x[9] + x[10] + x[11] + x[12] + x[13] + x[14] + x[15];      // c0
  coef[n] = c;
}

// ---------------------------------------------------------------------------
// Kernel 3: fused dual GEMM + epilogue; prev transposed in LDS, double-buffered.
// Wave w: M rows [w*16, w*16+16), all 128 N columns (8 sub-tiles).
// A-operand (16x4 f32): lanes 0-15 = M rows; lane half selects K+0 / K+2;
// VGPR pair holds K,K+1 (one b64 load). B-operand from transposed LDS tile:
// lane's (K,K+1) pair is contiguous -> ds_load_b64 (merged to ds_load_2addr).
// ---------------------------------------------------------------------------
__global__ __launch_bounds__(256) void dual_gemm_epilogue_kernel(
    const float* __restrict__ pa, const float* __restrict__ pb,
    const float* __restrict__ prev, const float4* __restrict__ coef,
    float* __restrict__ out) {
  __shared__ float sprev[2][BN * LSTR];  // 2 x 10 KB transposed prev chunks

  const int t = threadIdx.x;
  const int lane = t & 31;
  const int wave = t >> 5;
  const int mbase = blockIdx.y * BM + wave * 16;
  const int nbase = blockIdx.x * BN;

  const int l15 = lane & 15;
  const int mrow = mbase + l15;          // A-operand row for this lane
  const int khalf = (lane >> 4) << 1;    // 0 or 2: K offset by lane half

  const float* paRow = pa + (size_t)mrow * K_DIM;
  const float* pbRow = pb + (size_t)mrow * K_DIM;

  // Staging assignment: thread t covers n = t&127, k in {sk0..sk0+3, sk0+8..sk0+11}
  const int sn = t & 127;
  const int sk0 = (t >> 7) << 2;  // 0 or 4
  const float* sg = prev + nbase + sn;

  v8f accA[8] = {};
  v8f accB[8] = {};
  float4 r0, r1;  // prefetch registers for next chunk

  // ---- prologue: prefetch + stage chunk 0 into buffer 0 ----
  {
    const float* g = sg + (size_t)sk0 * N_DIM;
    r0.x = g[0 * N_DIM]; r0.y = g[1 * N_DIM];
    r0.z = g[2 * N_DIM]; r0.w = g[3 * N_DIM];
    const float* h = g + (size_t)8 * N_DIM;
    r1.x = h[0 * N_DIM]; r1.y = h[1 * N_DIM];
    r1.z = h[2 * N_DIM]; r1.w = h[3 * N_DIM];
    float* d = &sprev[0][sn * LSTR];
    *(float4*)(d + sk0) = r0;       // 16B-aligned ds_store_b128
    *(float4*)(d + sk0 + 8) = r1;
  }

  for (int c = 0; c < NCHUNK; ++c) {
    __syncthreads();  // staged data for chunk c visible; chunk c-1 readers done
    const int kc = c * KC;

    // prefetch chunk c+1 from global (latency hidden under compute below)
    if (c + 1 < NCHUNK) {
      const float* g = sg + (size_t)(kc + KC + sk0) * N_DIM;
      r0.x = g[0 * N_DIM]; r0.y = g[1 * N_DIM];
      r0.z = g[2 * N_DIM]; r0.w = g[3 * N_DIM];
      const float* h = g + (size_t)8 * N_DIM;
      r1.x = h[0 * N_DIM]; r1.y = h[1 * N_DIM];
      r1.z = h[2 * N_DIM]; r1.w = h[3 * N_DIM];
    }

    // compute: 4 K-steps x 8 N-subtiles x 2 matrices = 64 WMMA
    const float* sb = sprev[c & 1];
#pragma unroll
    for (int k0 = 0; k0 < KC; k0 += 4) {
      const int kl = k0 + khalf;            // local K for this lane half
      const v2f aA = *(const v2f*)(paRow + kc + kl);
      const v2f aB = *(const v2f*)(pbRow + kc + kl);
#pragma unroll
      for (int j = 0; j < 8; ++j) {
        const v2f bv = *(const v2f*)&sb[(j * 16 + l15) * LSTR + kl];
        accA[j] = __builtin_amdgcn_wmma_f32_16x16x4_f32(
            false, aA, false, bv, (short)0, accA[j], false, false);
        accB[j] = __builtin_amdgcn_wmma_f32_16x16x4_f32(
            false, aB, false, bv, (short)0, accB[j], false, false);
      }
    }

    // stage chunk c+1 into the other buffer (safe: its last readers were
    // chunk c-1, all past the barrier at the top of this iteration)
    if (c + 1 < NCHUNK) {
      float* d = &sprev[(c + 1) & 1][sn * LSTR];
      *(float4*)(d + sk0) = r0;
      *(float4*)(d + sk0 + 8) = r1;
    }
  }

  // ---- epilogue: C/D layout — VGPR r holds M = r (lanes 0-15) or r+8 ----
  const int half8 = (lane >> 4) << 3;
  float4 cf[8];
#pragma unroll
  for (int r = 0; r < 8; ++r) cf[r] = coef[mbase + half8 + r];

#pragma unroll
  for (int j = 0; j < 8; ++j) {
    const int n = nbase + j * 16 + l15;
#pragma unroll
    for (int r = 0; r < 8; ++r) {
      const float Av = accA[j][r];
      const float Bv = accB[j][r];
      const float o = cf[r].x + cf[r].y * Av + cf[r].z * Bv + cf[r].w * Av * Bv;
      out[(size_t)(mbase + half8 + r) * N_DIM + n] = o;
    }
  }
}

// ---------------------------------------------------------------------------
extern "C" void kernel_launch(void* const* d_in, const int* in_sizes, int n_in,
                              void* d_out, int out_size, void* d_ws,
                              size_t ws_size, hipStream_t stream) {
  const float* prev = (const float*)d_in[0];   // (1024, 8192)
  const float* wa   = (const float*)d_in[1];   // (4096, 1024)
  const float* wb   = (const float*)d_in[2];   // (4096, 1024)
  const float* tw   = (const float*)d_in[3];   // (16, 4096)
  float* out = (float*)d_out;                  // (4096, 8192)

  // Workspace: pa (16MB) | pb (16MB) | coef (64KB)
  float* pa = (float*)d_ws;
  float* pb = pa + (size_t)M_DIM * K_DIM;
  float4* coef = (float4*)(pb + (size_t)M_DIM * K_DIM);

  row_softmax_kernel<<<M_DIM, 256, 0, stream>>>(wa, pa);
  row_softmax_kernel<<<M_DIM, 256, 0, stream>>>(wb, pb);
  table_coeff_kernel<<<M_DIM / 256, 256, 0, stream>>>(tw, coef);

  dim3 grid(N_DIM / BN, M_DIM / BM);
  dual_gemm_epilogue_kernel<<<grid, 256, 0, stream>>>(pa, pb, prev, coef, out);
}